// MPNNet_v2_37323265803076
// MI455X (gfx1250) — compile-verified
//
#include <hip/hip_runtime.h>

#define DIM    64
#define NNODE  4096
#define NEDGE  8192
#define NB     128
#define NPG    32
#define NSTEM  512
#define FEAT   17
#define NSTEP  12
#define KDIM   128      // theta GEMM K
#define NCOL   4096     // theta GEMM N (DIM*DIM)
#define CAP    64       // per-node incoming-edge bucket capacity (mean deg = 2)

#define LDS_STRIDE 132  // 128 + 4 pad: addr/4 coefficient 33 (odd) spreads banks
#define THETA_LDS  ((64 * LDS_STRIDE + 128 * LDS_STRIDE) * 4)   // 101376 B

typedef __attribute__((ext_vector_type(2))) float v2f;
typedef __attribute__((ext_vector_type(4))) float v4f;
typedef __attribute__((ext_vector_type(8))) float v8f;

#if __has_builtin(__builtin_amdgcn_global_load_async_to_lds_b128) && \
    __has_builtin(__builtin_amdgcn_s_wait_asynccnt)
#define USE_ASYNC_LDS 1
typedef int v4i __attribute__((vector_size(16)));
typedef __attribute__((address_space(1))) v4i* g4p;   // global int4*
typedef __attribute__((address_space(3))) v4i* l4p;   // LDS int4*
__device__ __forceinline__ void async_copy16(const float* gsrc, float* ldst) {
    __builtin_amdgcn_global_load_async_to_lds_b128((g4p)const_cast<float*>(gsrc),
                                                   (l4p)ldst, 0, 0);
}
#else
#define USE_ASYNC_LDS 0
__device__ __forceinline__ void async_copy16(const float* gsrc, float* ldst) {
    *(v4f*)ldst = *(const v4f*)gsrc;
}
#endif

__device__ __forceinline__ float lrelu(float x) { return x > 0.f ? x : 0.01f * x; }
__device__ __forceinline__ float sigf(float x)  { return 1.f / (1.f + expf(-x)); }

// ---------------- lin0: out = lrelu(x @ W0^T + b0), [4096,17]->[4096,64] -----
__global__ void k_lin0(const float* __restrict__ x, const float* __restrict__ w,
                       const float* __restrict__ b, float* __restrict__ out) {
    int t = blockIdx.x * blockDim.x + threadIdx.x;   // 4096*64 threads
    int v = t >> 6, j = t & 63;
    float acc = b[j];
#pragma unroll
    for (int k = 0; k < FEAT; ++k) acc += x[v * FEAT + k] * w[j * FEAT + k];
    out[t] = lrelu(acc);
}

// ---------------- h1 = lrelu(edge_attr @ e1_w^T + e1_b), [8192,4]->[8192,128]
__global__ void k_h1(const float* __restrict__ ea, const float* __restrict__ w,
                     const float* __restrict__ b, float* __restrict__ h1) {
    int t = blockIdx.x * blockDim.x + threadIdx.x;   // 8192*128 threads
    int e = t >> 7, u = t & 127;
    float acc = b[u];
#pragma unroll
    for (int k = 0; k < 4; ++k) acc += ea[e * 4 + k] * w[u * 4 + k];
    h1[t] = lrelu(acc);
}

// ---------------- e2wT[k,n] = e2_w[n,k]  (one-time transpose, [128,4096]) ----
__global__ void k_tr(const float* __restrict__ e2w, float* __restrict__ e2t) {
    int t = blockIdx.x * blockDim.x + threadIdx.x;   // 128*4096 threads
    int n = t & 4095, k = t >> 12;
    e2t[k * NCOL + n] = e2w[n * KDIM + k];
}

// ---------------- degree / inv_deg / deterministic edge buckets -------------
__global__ void k_count(const int* __restrict__ dstA, int* __restrict__ deg) {
    int e = blockIdx.x * blockDim.x + threadIdx.x;
    if (e < NEDGE) atomicAdd(&deg[dstA[e]], 1);
}
__global__ void k_inv(const int* __restrict__ deg, float* __restrict__ invdeg) {
    int v = blockIdx.x * blockDim.x + threadIdx.x;
    if (v < NNODE) invdeg[v] = 1.f / fmaxf((float)deg[v], 1.f);
}
// stable rank of edge e among edges with same dst -> deterministic bucket fill
__global__ void k_rank(const int* __restrict__ dstA, int* __restrict__ eidb) {
    int e = blockIdx.x * blockDim.x + threadIdx.x;
    if (e >= NEDGE) return;
    int d = dstA[e];
    int r = 0;
    for (int e2 = 0; e2 < e; ++e2) r += (dstA[e2] == d) ? 1 : 0;
    if (r < CAP) eidb[d * CAP + r] = e;
}

// ---------------- theta GEMM via V_WMMA_F32_16X16X4_F32 ---------------------
// C[8192,4096] = A[8192,128] @ Bt[128,4096] + bias.
// Block = 64 rows x 128 cols. A(32KB)+B(64KB) tiles staged in LDS via
// GLOBAL_LOAD_ASYNC_TO_LDS_B128 (ASYNCcnt path). Each wave owns 16 cols and
// 4 M-tiles -> one B fetch feeds 4 WMMAs; 128 WMMAs per wave.
__global__ void k_theta(const float* __restrict__ A, const float* __restrict__ Bt,
                        const float* __restrict__ bias, float* __restrict__ C) {
    extern __shared__ float smem[];
    float* As = smem;                       // [64][LDS_STRIDE]
    float* Bs = smem + 64 * LDS_STRIDE;     // [128][LDS_STRIDE]

    int m0 = blockIdx.x * 64;
    int n0 = blockIdx.y * 128;
    int wave = threadIdx.x >> 5;
    int lane = threadIdx.x & 31;
    int mi = lane & 15;
    int g  = lane >> 4;

    // ---- stage A tile: rows m0..m0+63, K 0..127 (flat-contiguous in global)
#pragma unroll
    for (int i = 0; i < 8; ++i) {
        int idx4 = threadIdx.x + i * 256;            // 2048 4-float groups
        int r  = idx4 >> 5;
        int k4 = (idx4 & 31) * 4;
        async_copy16(A + (size_t)(m0 + r) * KDIM + k4, As + r * LDS_STRIDE + k4);
    }
    // ---- stage B tile: K 0..127, cols n0..n0+127
#pragma unroll
    for (int i = 0; i < 16; ++i) {
        int idx4 = threadIdx.x + i * 256;            // 4096 4-float groups
        int kk = idx4 >> 5;
        int c4 = (idx4 & 31) * 4;
        async_copy16(Bt + (size_t)kk * NCOL + n0 + c4, Bs + kk * LDS_STRIDE + c4);
    }
#if USE_ASYNC_LDS
    __builtin_amdgcn_s_wait_asynccnt(0);
#endif
    __syncthreads();

    int wcol = wave * 16 + mi;                        // this wave's column (B lane)
    v8f acc0 = {}, acc1 = {}, acc2 = {}, acc3 = {};
#pragma unroll
    for (int k = 0; k < KDIM; k += 4) {
        // B 4x16: lane halves split K: lanes 0-15 K={k,k+1}, lanes 16-31 K={k+2,k+3}
        v2f b;
        b[0] = Bs[(k + 2 * g + 0) * LDS_STRIDE + wcol];
        b[1] = Bs[(k + 2 * g + 1) * LDS_STRIDE + wcol];
        // A 16x4, same K split; 4 M-tiles share b
        v2f a0 = *(const v2f*)(As + (0 * 16 + mi) * LDS_STRIDE + k + 2 * g);
        v2f a1 = *(const v2f*)(As + (1 * 16 + mi) * LDS_STRIDE + k + 2 * g);
        v2f a2 = *(const v2f*)(As + (2 * 16 + mi) * LDS_STRIDE + k + 2 * g);
        v2f a3 = *(const v2f*)(As + (3 * 16 + mi) * LDS_STRIDE + k + 2 * g);
        acc0 = __builtin_amdgcn_wmma_f32_16x16x4_f32(false, a0, false, b, (short)0, acc0, false, false);
        acc1 = __builtin_amdgcn_wmma_f32_16x16x4_f32(false, a1, false, b, (short)0, acc1, false, false);
        acc2 = __builtin_amdgcn_wmma_f32_16x16x4_f32(false, a2, false, b, (short)0, acc2, false, false);
        acc3 = __builtin_amdgcn_wmma_f32_16x16x4_f32(false, a3, false, b, (short)0, acc3, false, false);
    }

    // C/D: lane half g selects rows 8g..8g+7 within a tile, column = lane&15
    int col = n0 + wcol;
    float bv = bias[col];
    v8f accs[4] = {acc0, acc1, acc2, acc3};
#pragma unroll
    for (int mt = 0; mt < 4; ++mt) {
        int row0 = m0 + mt * 16 + g * 8;
#pragma unroll
        for (int r = 0; r < 8; ++r)
            C[(size_t)(row0 + r) * NCOL + col] = accs[mt][r] + bv;
    }
}

// ---------------- msg[e,:] = out[src[e],:] (1x64) @ theta[e] (64x64) --------
// one wave32 per edge; lane owns 2 output cols; __shfl broadcasts src row.
__global__ void k_msg(const float* __restrict__ out, const float* __restrict__ theta,
                      const int* __restrict__ srcA, float* __restrict__ msg) {
    int wave = threadIdx.x >> 5, lane = threadIdx.x & 31;
    int e = blockIdx.x * 8 + wave;
    int s = srcA[e];
    float r0 = out[s * 64 + lane];
    float r1 = out[s * 64 + 32 + lane];
    const float* th = theta + (size_t)e * NCOL + 2 * lane;
    __builtin_prefetch(th, 0, 1);            // global_prefetch_b8 warm-up
    float a0 = 0.f, a1 = 0.f;
#pragma unroll 8
    for (int i = 0; i < 32; ++i) {
        float v = __shfl(r0, i, 32);
        v2f tv = *(const v2f*)(th + i * 64);
        a0 += v * tv[0];
        a1 += v * tv[1];
    }
#pragma unroll 8
    for (int i = 0; i < 32; ++i) {
        float v = __shfl(r1, i, 32);
        v2f tv = *(const v2f*)(th + (32 + i) * 64);
        a0 += v * tv[0];
        a1 += v * tv[1];
    }
    msg[e * 64 + 2 * lane]     = a0;
    msg[e * 64 + 2 * lane + 1] = a1;
}

// ---------------- m = lrelu(agg*inv_deg + out@root + conv_b) ----------------
__global__ void k_aggm(const float* __restrict__ out, const float* __restrict__ msg,
                       const int* __restrict__ deg, const float* __restrict__ invdeg,
                       const int* __restrict__ eidb, const float* __restrict__ root,
                       const float* __restrict__ cb, float* __restrict__ m) {
    int t = blockIdx.x * blockDim.x + threadIdx.x;   // 4096*64
    int v = t >> 6, j = t & 63;
    int cnt = deg[v]; if (cnt > CAP) cnt = CAP;
    float s = 0.f;
    for (int p = 0; p < cnt; ++p) s += msg[eidb[v * CAP + p] * 64 + j];
    s *= invdeg[v];
    const float* ov = out + v * 64;
#pragma unroll 16
    for (int k = 0; k < 64; ++k) s += ov[k] * root[k * 64 + j];
    m[t] = lrelu(s + cb[j]);
}

// ---------------- GRU cell: hout = GRU(m, h) --------------------------------
__global__ void k_gru(const float* __restrict__ m, const float* __restrict__ h,
                      const float* __restrict__ wih, const float* __restrict__ whh,
                      const float* __restrict__ bih, const float* __restrict__ bhh,
                      float* __restrict__ hout) {
    int t = blockIdx.x * blockDim.x + threadIdx.x;   // 4096*64
    int v = t >> 6, j = t & 63;
    const float* mv = m + v * 64;
    const float* hv = h + v * 64;
    float ir = bih[j], iz = bih[64 + j], in = bih[128 + j];
    float hr = bhh[j], hz = bhh[64 + j], hn = bhh[128 + j];
#pragma unroll 8
    for (int k = 0; k < 64; ++k) {
        float mk = mv[k], hk = hv[k];
        ir += mk * wih[j * 64 + k];
        iz += mk * wih[(64 + j) * 64 + k];
        in += mk * wih[(128 + j) * 64 + k];
        hr += hk * whh[j * 64 + k];
        hz += hk * whh[(64 + j) * 64 + k];
        hn += hk * whh[(128 + j) * 64 + k];
    }
    float r = sigf(ir + hr);
    float z = sigf(iz + hz);
    float n = tanhf(in + r * hn);
    hout[t] = (1.f - z) * n + z * hv[j];
}

// ---------------- stem head -------------------------------------------------
__global__ void k_stem1(const float* __restrict__ out, const int* __restrict__ ns,
                        const int* __restrict__ st, const int* __restrict__ sb,
                        const float* __restrict__ w1, const float* __restrict__ b1,
                        float* __restrict__ tbuf) {
    int t = blockIdx.x * blockDim.x + threadIdx.x;   // 512*512
    int s = t >> 9, u = t & 511;
    int node = ns[sb[s]] + st[s];
    const float* ov = out + node * 64;
    float acc = b1[u];
#pragma unroll 8
    for (int k = 0; k < 64; ++k) acc += ov[k] * w1[u * 64 + k];
    tbuf[t] = lrelu(acc);
}
__global__ void k_stem2(const float* __restrict__ tbuf, const float* __restrict__ w2,
                        const float* __restrict__ b2, float* __restrict__ per_stem) {
    int t = blockIdx.x * blockDim.x + threadIdx.x;   // 512*105
    if (t >= NSTEM * 105) return;
    int s = t / 105, c = t % 105;
    float acc = b2[c];
    const float* tv = tbuf + s * 512;
#pragma unroll 8
    for (int u = 0; u < 512; ++u) acc += tv[u] * w2[c * 512 + u];
    per_stem[t] = acc;
}

// ---------------- Set2Set (3 LSTM+attention iters) + lin3, one block/graph --
__global__ void k_s2s(const float* __restrict__ out,
                      const float* __restrict__ wih, const float* __restrict__ whh,
                      const float* __restrict__ bih, const float* __restrict__ bhh,
                      const float* __restrict__ w3, const float* __restrict__ b3,
                      float* __restrict__ per_mol) {
    __shared__ float q[128], hh[64], cc[64], gates[256], ebuf[32], abuf[32], inv_sum;
    int b = blockIdx.x, t = threadIdx.x;
    if (t < 128) q[t] = 0.f;
    if (t < 64) { hh[t] = 0.f; cc[t] = 0.f; }
    __syncthreads();
    for (int it = 0; it < 3; ++it) {
        // LSTM gate pre-activations: 256 threads, one gate each
        float g = bih[t] + bhh[t];
        for (int k = 0; k < 128; ++k) g += q[k] * wih[t * 128 + k];
        for (int k = 0; k < 64;  ++k) g += hh[k] * whh[t * 64 + k];
        gates[t] = g;
        __syncthreads();
        if (t < 64) {
            float i  = sigf(gates[t]),        f = sigf(gates[64 + t]);
            float gg = tanhf(gates[128 + t]), o = sigf(gates[192 + t]);
            float c = f * cc[t] + i * gg;
            cc[t] = c;
            hh[t] = o * tanhf(c);
        }
        __syncthreads();
        if (t < 32) {                        // attention logits over 32 nodes
            const float* ov = out + (b * NPG + t) * 64;
            float e = 0.f;
#pragma unroll 8
            for (int k = 0; k < 64; ++k) e += ov[k] * hh[k];
            ebuf[t] = e;
        }
        __syncthreads();
        if (t == 0) {                        // softmax normalizer
            float mx = ebuf[0];
            for (int j = 1; j < NPG; ++j) mx = fmaxf(mx, ebuf[j]);
            float ssum = 0.f;
            for (int j = 0; j < NPG; ++j) { float a = expf(ebuf[j] - mx); abuf[j] = a; ssum += a; }
            inv_sum = 1.f / ssum;
        }
        __syncthreads();
        if (t < 64) {                        // r = sum_j a_j * out_j ; q* = [hh, r]
            float r = 0.f;
            for (int j = 0; j < NPG; ++j) r += abuf[j] * out[(b * NPG + j) * 64 + t];
            r *= inv_sum;
            q[t] = hh[t];
            q[64 + t] = r;
        }
        __syncthreads();
    }
    if (t == 0) {
        float acc = b3[0];
        for (int k = 0; k < 128; ++k) acc += q[k] * w3[k];
        per_mol[b] = acc;
    }
}

extern "C" void kernel_launch(void* const* d_in, const int* in_sizes, int n_in,
                              void* d_out, int out_size, void* d_ws, size_t ws_size,
                              hipStream_t stream) {
    (void)in_sizes; (void)n_in; (void)out_size; (void)ws_size;
    const float* x     = (const float*)d_in[0];
    const int*   eidx  = (const int*)d_in[1];
    const int*   srcA  = eidx;
    const int*   dstA  = eidx + NEDGE;
    const float* ea    = (const float*)d_in[2];
    const int*   ns    = (const int*)d_in[4];
    const int*   st    = (const int*)d_in[5];
    const int*   sb    = (const int*)d_in[6];
    const float* l0w   = (const float*)d_in[7];
    const float* l0b   = (const float*)d_in[8];
    const float* e1w   = (const float*)d_in[9];
    const float* e1b   = (const float*)d_in[10];
    const float* e2w   = (const float*)d_in[11];
    const float* e2b   = (const float*)d_in[12];
    const float* root  = (const float*)d_in[13];
    const float* cb    = (const float*)d_in[14];
    const float* gwih  = (const float*)d_in[15];
    const float* gwhh  = (const float*)d_in[16];
    const float* gbih  = (const float*)d_in[17];
    const float* gbhh  = (const float*)d_in[18];
    const float* lwih  = (const float*)d_in[19];
    const float* lwhh  = (const float*)d_in[20];
    const float* lbih  = (const float*)d_in[21];
    const float* lbhh  = (const float*)d_in[22];
    const float* w1    = (const float*)d_in[23];
    const float* b1    = (const float*)d_in[24];
    const float* w2    = (const float*)d_in[25];
    const float* b2    = (const float*)d_in[26];
    const float* w3    = (const float*)d_in[27];
    const float* b3    = (const float*)d_in[28];

    char* wp = (char*)d_ws;
    auto alloc = [&](size_t bytes) { void* p = (void*)wp;
        wp += (bytes + 255) & ~(size_t)255; return p; };
    float* theta  = (float*)alloc((size_t)NEDGE * NCOL * 4);   // 134 MB, L2-resident
    float* h1     = (float*)alloc((size_t)NEDGE * KDIM * 4);
    float* e2t    = (float*)alloc((size_t)KDIM * NCOL * 4);
    float* bufA   = (float*)alloc((size_t)NNODE * DIM * 4);
    float* bufB   = (float*)alloc((size_t)NNODE * DIM * 4);
    float* mbuf   = (float*)alloc((size_t)NNODE * DIM * 4);
    float* msg    = (float*)alloc((size_t)NEDGE * DIM * 4);
    int*   deg    = (int*)  alloc((size_t)NNODE * 4);
    float* invdeg = (float*)alloc((size_t)NNODE * 4);
    int*   eidb   = (int*)  alloc((size_t)NNODE * CAP * 4);
    float* tstem  = (float*)alloc((size_t)NSTEM * 512 * 4);

    float* per_stem = (float*)d_out;            // 512*105
    float* per_mol  = (float*)d_out + NSTEM * 105;

    // setup
    k_lin0 <<<NNODE * DIM / 256, 256, 0, stream>>>(x, l0w, l0b, bufA);
    k_h1   <<<NEDGE * KDIM / 256, 256, 0, stream>>>(ea, e1w, e1b, h1);
    k_tr   <<<KDIM * NCOL / 256, 256, 0, stream>>>(e2w, e2t);
    (void)hipMemsetAsync(deg, 0, (size_t)NNODE * 4, stream);
    k_count<<<NEDGE / 256, 256, 0, stream>>>(dstA, deg);
    k_inv  <<<NNODE / 256, 256, 0, stream>>>(deg, invdeg);
    k_rank <<<NEDGE / 256, 256, 0, stream>>>(dstA, eidb);

    // theta = h1 @ e2_w^T + e2_b  (fp32 WMMA, LDS-tiled 64x128 blocks)
    dim3 tgrid(NEDGE / 64, NCOL / 128);
    k_theta<<<tgrid, 256, THETA_LDS, stream>>>(h1, e2t, e2b, theta);

    // 12 conv + GRU steps (ping-pong node state)
    float* cur = bufA;
    float* nxt = bufB;
    for (int s = 0; s < NSTEP; ++s) {
        k_msg <<<NEDGE / 8, 256, 0, stream>>>(cur, theta, srcA, msg);
        k_aggm<<<NNODE * DIM / 256, 256, 0, stream>>>(cur, msg, deg, invdeg,
                                                      eidb, root, cb, mbuf);
        k_gru <<<NNODE * DIM / 256, 256, 0, stream>>>(mbuf, cur, gwih, gwhh,
                                                      gbih, gbhh, nxt);
        float* tmp = cur; cur = nxt; nxt = tmp;
    }

    // heads
    k_stem1<<<NSTEM * 512 / 256, 256, 0, stream>>>(cur, ns, st, sb, w1, b1, tstem);
    k_stem2<<<(NSTEM * 105 + 255) / 256, 256, 0, stream>>>(tstem, w2, b2, per_stem);
    k_s2s  <<<NB, 256, 0, stream>>>(cur, lwih, lwhh, lbih, lbhh, w3, b3, per_mol);
}